// HIRNNLayer_41721312314378
// MI455X (gfx1250) — compile-verified
//
#include <hip/hip_runtime.h>
#include <hip/hip_bf16.h>
#include <stdint.h>

// ---------------------------------------------------------------------------
// HIRNN hydrological scan for gfx1250 (MI455X).
//  - One wave32 per block, one batch-chain per lane (B=4096 -> 128 blocks).
//  - TDM (tensor_load_to_lds) double-buffered staging of the strided input
//    tile (32 rows x 64 floats, row stride 2048 floats) with LDS padding so
//    lane reads are bank-conflict free. Synced via s_wait_tensorcnt.
//  - All heaviside() constructs reduced exactly to min/max; single v_exp_f32
//    per step; scan fused with the post-roll flux recomputation (Q[t>=1] uses
//    the pre-step state == rolled state; Q[0] patched with the final state).
// ---------------------------------------------------------------------------

#define T_LEN   1024
#define CHUNK_T 32                    // timesteps per TDM tile
#define NCHUNK  (T_LEN / CHUNK_T)     // 32 chunks
#define ROW_DW  (CHUNK_T * 2 + 2)     // 64 data dwords + 2 pad dwords per row
#define BUF_DW  (32 * ROW_DW)         // one buffer: 32 rows

#if defined(__HIP_DEVICE_COMPILE__) && \
    __has_builtin(__builtin_amdgcn_tensor_load_to_lds) && \
    __has_builtin(__builtin_amdgcn_s_wait_tensorcnt)
#define HAVE_TDM 1
#else
#define HAVE_TDM 0
#endif

typedef unsigned int v4u __attribute__((ext_vector_type(4)));
typedef int          v4i __attribute__((ext_vector_type(4)));
typedef int          v8i __attribute__((ext_vector_type(8)));

#if HAVE_TDM
__device__ __forceinline__ void tdm_load_tile(v4u g0, v8i g1) {
  v4i z4 = {0, 0, 0, 0};
#if defined(__clang_major__) && (__clang_major__ >= 23)
  v8i z8 = {0, 0, 0, 0, 0, 0, 0, 0};
  __builtin_amdgcn_tensor_load_to_lds(g0, g1, z4, z4, z8, 0);
#else
  __builtin_amdgcn_tensor_load_to_lds(g0, g1, z4, z4, 0);
#endif
}
#endif

// One fused timestep: returns Q for this step's (pre-update) state and
// advances (S_sms, S_gw) in place. Heavisides reduced to exact min/max forms.
__device__ __forceinline__ float hydro_step(
    float Prec, float PET,
    float cINSC, float cCOEFF, float cSMSC, float invSMSC, float negSQdS,
    float cSUB, float cCRAK, float cRecK,
    float& S_sms, float& S_gw)
{
  float IMAX = fmaxf(fminf(cINSC, PET), 0.0f);
  float INTc = fmaxf(fminf(IMAX, Prec), 0.0f);
  float INR  = fmaxf(Prec - INTc, 0.0f);
  float SMSc = fmaxf(fminf(S_sms, cSMSC), 0.0f);
  float frac = SMSc * invSMSC;
  float cap  = cCOEFF * __expf(negSQdS * SMSc);
  // heaviside(cap-INR)*INR + heaviside(INR-cap)*cap  ==  min(INR, cap)
  float RMO  = fmaxf(fminf(cap, INR), 0.0f);
  float IRUN = fmaxf(INR - RMO, 0.0f);
  float SRUN = fmaxf(cSUB * frac * RMO, 0.0f);
  float REC  = fmaxf(cCRAK * frac * (RMO - SRUN), 0.0f);
  float SMF  = fmaxf(RMO - SRUN - REC, 0.0f);
  float POT  = fmaxf(PET - INTc, 0.0f);
  float rr   = 10.0f * frac;
  float ETS  = fmaxf(fminf(POT, rr), 0.0f);              // min(POT, r)
  float BAS  = fmaxf(cRecK * S_gw, 0.0f);                // heaviside(GW)*K*GW
  float Q    = fmaxf(fmaxf(SRUN + IRUN, 0.0f) + BAS, 0.0f);
  float d    = SMF - ETS;
  float s_   = SMSc + d;
  // heaviside(s-SMSC)*(REC+s-SMSC) + heaviside(SMSC-s)*REC == REC+max(s-SMSC,0)
  float RECn = fmaxf(REC + fmaxf(s_ - cSMSC, 0.0f), 0.0f);
  S_sms = SMSc + fminf(fmaxf(d, -100000.0f), 100000.0f);
  S_gw  = S_gw + fminf(fmaxf(RECn - BAS, -100000.0f), 100000.0f);
  return Q;
}

__global__ __launch_bounds__(32) void hydro_scan_kernel(
    const float* __restrict__ in,     // (B, T, 2) interleaved Prec, PET
    const float* __restrict__ pINSC, const float* __restrict__ pCOEFF,
    const float* __restrict__ pSQ,   const float* __restrict__ pSMSC,
    const float* __restrict__ pSUB,  const float* __restrict__ pCRAK,
    const float* __restrict__ pRecK,
    float* __restrict__ out)          // (B, T)
{
  const int lane  = threadIdx.x;            // 0..31, one chain per lane
  const int bbase = blockIdx.x * 32;
  const int b     = bbase + lane;

  // Loop-invariant clipped parameters (uniform -> SALU/SGPR).
  const float cINSC  = fminf(fmaxf(pINSC[0]  * 5.0f,   0.5f),   5.0f);
  const float cCOEFF = fminf(fmaxf(pCOEFF[0] * 400.0f, 50.0f),  400.0f);
  const float cSQ    = fminf(fmaxf(pSQ[0]    * 6.0f,   0.0f),   6.0f);
  const float cSMSC  = fminf(fmaxf(pSMSC[0]  * 500.0f, 50.0f),  500.0f);
  const float cSUB   = fminf(fmaxf(pSUB[0],            0.0f),   1.0f);
  const float cCRAK  = fminf(fmaxf(pCRAK[0],           0.0f),   1.0f);
  const float cRecK  = fminf(fmaxf(pRecK[0]  * 0.3f,   0.003f), 0.3f);
  const float invSMSC = 1.0f / cSMSC;
  const float negSQdS = -cSQ * invSMSC;

  float S_sms = 0.0f, S_gw = 0.0f;
  float Prec0 = 0.0f, PET0 = 0.0f;
  float* __restrict__ outb = out + (size_t)b * T_LEN;

#if HAVE_TDM
  __shared__ float sh[2][BUF_DW];
  const unsigned ldsOff[2] = {
      (unsigned)(uintptr_t)&sh[0][0],
      (unsigned)(uintptr_t)&sh[1][0]
  };
  // D# group1 (constant): mask=0, data_size=4B, pad_enable, pad_interval=64dw,
  // pad_amount=2dw, tensor_dim0=2048, tensor_dim1=4096, tile=64x32,
  // tensor_dim0_stride=2048 elements.
  const v8i g1 = { (int)0x03520000, (int)0x08000000, (int)0x10000000,
                   (int)0x00400000, 32, 2048, 0, 0 };
  const uint64_t gbase =
      (uint64_t)(uintptr_t)in + (uint64_t)bbase * (T_LEN * 2u * 4u);

  auto issue_chunk = [&](int chunk) {
    uint64_t ga = gbase + (uint64_t)chunk * (CHUNK_T * 2u * 4u);
    v4u g0;
    g0.x = 1u;                                        // count=1, user D#
    g0.y = ldsOff[chunk & 1];                         // LDS byte address
    g0.z = (unsigned)(ga & 0xFFFFFFFFu);              // global_addr[31:0]
    g0.w = (unsigned)((ga >> 32) & 0x01FFFFFFu)       // global_addr[56:32]
         | 0x80000000u;                               // type=2 ("image")
    tdm_load_tile(g0, g1);
  };

  issue_chunk(0);
#endif

  for (int chunk = 0; chunk < NCHUNK; ++chunk) {
#if HAVE_TDM
    if (chunk + 1 < NCHUNK) {
      issue_chunk(chunk + 1);                 // prefetch next tile (other buf)
      __builtin_amdgcn_s_wait_tensorcnt(1);   // this tile done (in-order)
    } else {
      __builtin_amdgcn_s_wait_tensorcnt(0);
    }
    const float* __restrict__ row = &sh[chunk & 1][lane * ROW_DW];
#else
    const float* __restrict__ row =
        in + ((size_t)b * T_LEN + (size_t)chunk * CHUNK_T) * 2;
    __builtin_prefetch(row + 2 * CHUNK_T, 0, 1);  // global_prefetch_b8
#endif

#pragma unroll
    for (int tt = 0; tt < CHUNK_T; ++tt) {
      const float2 xv = *reinterpret_cast<const float2*>(&row[2 * tt]);
      if (chunk == 0 && tt == 0) { Prec0 = xv.x; PET0 = xv.y; }
      float Q = hydro_step(xv.x, xv.y, cINSC, cCOEFF, cSMSC, invSMSC, negSQdS,
                           cSUB, cCRAK, cRecK, S_sms, S_gw);
      int t = chunk * CHUNK_T + tt;
      if (t > 0) outb[t] = Q;   // Q[t>=1] uses pre-step (== rolled) state
    }
  }

  // roll(shift=1): Q[0] is computed from the FINAL state with x[0].
  {
    float s0 = S_sms, g0s = S_gw;
    float Q0 = hydro_step(Prec0, PET0, cINSC, cCOEFF, cSMSC, invSMSC, negSQdS,
                          cSUB, cCRAK, cRecK, s0, g0s);
    outb[0] = Q0;
  }
}

extern "C" void kernel_launch(void* const* d_in, const int* in_sizes, int n_in,
                              void* d_out, int out_size, void* d_ws, size_t ws_size,
                              hipStream_t stream) {
  (void)n_in; (void)out_size; (void)d_ws; (void)ws_size;
  const float* in    = (const float*)d_in[0];
  const float* INSC  = (const float*)d_in[1];
  const float* COEFF = (const float*)d_in[2];
  const float* SQ    = (const float*)d_in[3];
  const float* SMSC  = (const float*)d_in[4];
  const float* SUB   = (const float*)d_in[5];
  const float* CRAK  = (const float*)d_in[6];
  const float* RecK  = (const float*)d_in[7];
  float* out = (float*)d_out;

  const int B = in_sizes[0] / (T_LEN * 2);   // 4096
  hydro_scan_kernel<<<dim3(B / 32), dim3(32), 0, stream>>>(
      in, INSC, COEFF, SQ, SMSC, SUB, CRAK, RecK, out);
}